// deephi_Correlation2D_Elemwise_43963285242021
// MI455X (gfx1250) — compile-verified
//
#include <hip/hip_runtime.h>

// Problem constants (static per the reference: B=4, C=32, H=64, W=128, p=4)
#define BB 4
#define CC 32
#define HH 64
#define WW 128
#define PP 4
#define DD 9            // 2*p+1
#define NK 81           // D*D shifts
#define S2W (WW + 2*PP) // 136 padded row width
#define S2ELEMS (DD * S2W)

typedef float floatx4 __attribute__((ext_vector_type(4)));

__global__ __launch_bounds__(256) void corr2d_costvol_kernel(
    const float* __restrict__ in1,
    const float* __restrict__ in2,
    float* __restrict__ out)
{
    // LDS staging: one in1 row + 9-row zero-padded in2 halo tile
    __shared__ float s1[WW];
    __shared__ float s2[DD][S2W];

    const int t    = threadIdx.x;
    const int bc_h = blockIdx.x;        // 0 .. B*C*H-1
    const int h    = bc_h % HH;
    const int bc   = bc_h / HH;         // b*C + c

    const float* in1_row  = in1 + (size_t)bc * (HH * WW) + (size_t)h * WW;
    const float* in2_base = in2 + (size_t)bc * (HH * WW);

    // ---- Stage in1 row via async global->LDS (ASYNCcnt path, CDNA5) ----
    if (t < WW) {
        unsigned laddr = (unsigned)(unsigned long long)(&s1[t]);
        const float* g = in1_row + t;
        asm volatile("global_load_async_to_lds_b32 %0, %1, off"
                     :: "v"(laddr), "v"(g) : "memory");
    }

    // ---- Stage 9x136 in2 halo tile: async loads for in-range, ds-store zeros
    //      for padding (disjoint addresses, so no async/ds ordering hazard) ----
    for (int idx = t; idx < S2ELEMS; idx += 256) {
        const int r  = idx / S2W;       // 0..8  -> input row h + r - 4
        const int c  = idx - r * S2W;   // 0..135 -> input col c - 4
        const int hh = h + r - PP;
        const int ww = c - PP;
        if (hh >= 0 && hh < HH && ww >= 0 && ww < WW) {
            unsigned laddr = (unsigned)(unsigned long long)(&s2[r][c]);
            const float* g = in2_base + hh * WW + ww;
            asm volatile("global_load_async_to_lds_b32 %0, %1, off"
                         :: "v"(laddr), "v"(g) : "memory");
        } else {
            s2[r][c] = 0.0f;
        }
    }

    // Drain this wave's async copies, then sync the workgroup.
    asm volatile("s_wait_asynccnt 0" ::: "memory");
    __syncthreads();

    // ---- Compute + stream out: wave w handles shifts k = w, w+8, ... ----
    const int wave = t >> 5;
    const int lane = t & 31;
    const int w4   = lane << 2;         // 4 consecutive outputs per lane

    // in1 operand is invariant across all 81 shifts: hoist to registers.
    const float a0 = s1[w4 + 0];
    const float a1 = s1[w4 + 1];
    const float a2 = s1[w4 + 2];
    const float a3 = s1[w4 + 3];

    // out[((bc*81 + k)*H + h)*W + w4]
    const size_t out_base = ((size_t)bc * NK) * (size_t)(HH * WW)
                          + (size_t)h * WW + (size_t)w4;

    for (int k = wave; k < NK; k += 8) {
        const int di = k / DD;
        const int dj = k - di * DD;

        const float* s2p = &s2[di][w4 + dj];
        const float b0 = s2p[0];
        const float b1 = s2p[1];
        const float b2 = s2p[2];
        const float b3 = s2p[3];

        floatx4 v;
        v.x = a0 * b0;
        v.y = a1 * b1;
        v.z = a2 * b2;
        v.w = a3 * b3;

        // 16B-aligned, lane-contiguous (512B/wave) non-temporal stream store.
        float* op = out + out_base + (size_t)k * (size_t)(HH * WW);
        __builtin_nontemporal_store(v, (floatx4*)op);
    }
}

extern "C" void kernel_launch(void* const* d_in, const int* in_sizes, int n_in,
                              void* d_out, int out_size, void* d_ws, size_t ws_size,
                              hipStream_t stream) {
    const float* in1 = (const float*)d_in[0];
    const float* in2 = (const float*)d_in[1];
    // d_in[2] is pad_size (==4), static in this build.
    float* out = (float*)d_out;

    dim3 grid(BB * CC * HH);  // 8192 workgroups, one per (b,c,h) row
    dim3 block(256);          // 8 wave32 waves
    corr2d_costvol_kernel<<<grid, block, 0, stream>>>(in1, in2, out);
}